// gru_dual_stage_attention_81595788689582
// MI455X (gfx1250) — compile-verified
//
#include <hip/hip_runtime.h>

typedef __attribute__((ext_vector_type(2))) float v2f;
typedef __attribute__((ext_vector_type(8))) float v8f;

#define NB    8192
#define NT    24
#define NENC  25     // h0 + 24 encoder states

__device__ __forceinline__ v2f mk2(float a, float b) { v2f r; r.x = a; r.y = b; return r; }

__device__ __forceinline__ v8f bc8(float v) {
  v8f r;
#pragma unroll
  for (int p = 0; p < 8; ++p) r[p] = v;
  return r;
}

__device__ __forceinline__ v8f wmma4(v2f a, v2f b, v8f c) {
  // D = A(16x4 f32) * B(4x16 f32) + C(16x16 f32)
  return __builtin_amdgcn_wmma_f32_16x16x4_f32(false, a, false, b, (short)0, c, false, false);
}

// raw v_rcp_f32 / v_exp_f32 (avoid IEEE div_scale/div_fmas sequences)
__device__ __forceinline__ float frcp(float x)  { return __builtin_amdgcn_rcpf(x); }
__device__ __forceinline__ float fexp2(float x) { return __builtin_amdgcn_exp2f(x); }

__device__ __forceinline__ float sigm(float x) {
  // rcp(1 + 2^(-x*log2e)); inf-safe: x<<0 -> rcp(inf)=0, x>>0 -> rcp(1)=1
  return frcp(1.f + fexp2(-1.44269504f * x));
}

__device__ __forceinline__ float ftanh(float x) {
#if __has_builtin(__builtin_amdgcn_tanhf)
  return __builtin_amdgcn_tanhf(x);
#elif __has_builtin(__builtin_amdgcn_tanh_f32)
  return __builtin_amdgcn_tanh_f32(x);
#else
  // (t-1)/(t+1), t = e^(2x) = 2^(2x*log2e); clamp keeps t finite (tanh(9)==1 in fp32)
  const float cx = fminf(9.f, fmaxf(-9.f, x));
  const float t  = fexp2(2.88539009f * cx);
  return (t - 1.f) * frcp(t + 1.f);
#endif
}

__global__ __launch_bounds__(32) void darnn_kernel(
    const float* __restrict__ x,    const float* __restrict__ h0e, const float* __restrict__ h0d,
    const float* __restrict__ a1w,  const float* __restrict__ a1b,
    const float* __restrict__ a2w,  const float* __restrict__ a2b,
    const float* __restrict__ a3w,  const float* __restrict__ a3b,
    const float* __restrict__ ewih, const float* __restrict__ ewhh,
    const float* __restrict__ ebih, const float* __restrict__ ebhh,
    const float* __restrict__ l1w,  const float* __restrict__ l1b,
    const float* __restrict__ l2w,  const float* __restrict__ l2b,
    const float* __restrict__ l3w,  const float* __restrict__ l3b,
    const float* __restrict__ l4w,  const float* __restrict__ l4b,
    const float* __restrict__ dwih, const float* __restrict__ dwhh,
    const float* __restrict__ dbih, const float* __restrict__ dbhh,
    const float* __restrict__ fcw,  const float* __restrict__ fcb,
    float* __restrict__ out)
{
  // Padded LDS tiles (even strides so A-fragment chunks are ds_load_b64-aligned)
  __shared__ float sENC[NENC * 16 * 18];  // encoder states, row-major 16x16, stride 18
  __shared__ float sHD[16 * 34];          // decoder hidden 16x32, stride 34
  __shared__ float sYT[16 * 34];          // y0 (GRU input) 16x32, stride 34
  __shared__ float sY0[16 * 22];          // [wc | x | 0] 16x20, stride 22
  __shared__ float sAT[16 * 25];          // attention scores 16x24, stride 25
  __shared__ float sL4W[32 * 22];         // l4_w zero-padded K: 19 -> 22

  const int lane = threadIdx.x;
  const int n    = lane & 15;   // fragment column / batch row for per-lane work
  const int hi   = lane >> 4;   // lane half
  const int base = blockIdx.x * 16;
  const int b    = base + n;

  // ---- cooperative LDS fill: padded l4_w (32 rows, K=19 padded to 22 w/ zeros)
  {
    const int r = lane; // 32 lanes == 32 rows
#pragma unroll
    for (int k = 0; k < 19; ++k) sL4W[r * 22 + k] = l4w[r * 19 + k];
    sL4W[r * 22 + 19] = 0.f; sL4W[r * 22 + 20] = 0.f; sL4W[r * 22 + 21] = 0.f;
  }

  // ---- preloaded B-fragments (per-lane v2f; B[k][n] = W[n_total][k])
  v2f fEIH[3], fEHH[3][4], fL1[2][8], fL2[2][4];
#pragma unroll
  for (int g = 0; g < 3; ++g) {
    const int row = g * 16 + n;
    fEIH[g] = hi ? mk2(ewih[row * 3 + 2], 0.f)
                 : mk2(ewih[row * 3 + 0], ewih[row * 3 + 1]);
#pragma unroll
    for (int c = 0; c < 4; ++c)
      fEHH[g][c] = *(const v2f*)&ewhh[row * 16 + 4 * c + 2 * hi];
  }
#pragma unroll
  for (int g = 0; g < 2; ++g) {
#pragma unroll
    for (int c = 0; c < 8; ++c) fL1[g][c] = *(const v2f*)&l1w[(g * 16 + n) * 32 + 4 * c + 2 * hi];
#pragma unroll
    for (int c = 0; c < 4; ++c) fL2[g][c] = *(const v2f*)&l2w[(g * 16 + n) * 16 + 4 * c + 2 * hi];
  }

  // ---- per-lane bias scalars (bias depends only on fragment column n)
  const float b_r  = ebih[n] + ebhh[n];
  const float b_z  = ebih[16 + n] + ebhh[16 + n];
  const float b_ic = ebih[32 + n];
  const float b_hc = ebhh[32 + n];
  const float bl1a = l1b[n],      bl1b = l1b[16 + n];
  const float bl2a = l2b[n],      bl2b = l2b[16 + n];
  const float bl4a = l4b[n],      bl4b = l4b[16 + n];
  const float bdr0 = dbih[n] + dbhh[n],           bdr1 = dbih[16 + n] + dbhh[16 + n];
  const float bdz0 = dbih[32 + n] + dbhh[32 + n], bdz1 = dbih[48 + n] + dbhh[48 + n];
  const float bdi0 = dbih[64 + n], bdi1 = dbih[80 + n];
  const float bdh0 = dbhh[64 + n], bdh1 = dbhh[80 + n];
  const float l3wa = l3w[n], l3wb = l3w[16 + n], l3bs = l3b[0];

  // ---- initial states (C-fragment layout: vgpr p -> row p+8*hi, lane -> col n)
  v8f hC, hd0, hd1;
#pragma unroll
  for (int p = 0; p < 8; ++p) {
    const int m = base + p + 8 * hi;
    const float v = h0e[m * 16 + n];
    hC[p] = v;
    sENC[(p + 8 * hi) * 18 + n] = v;  // ENC[0] = h0
    hd0[p] = h0d[m * 32 + n];
    hd1[p] = h0d[m * 32 + 16 + n];
  }
  __syncthreads();

  // ============================ ENCODER ============================
#pragma unroll 1
  for (int t = 0; t < NT; ++t) {
    const float* encT = &sENC[t * 288];
    float hrow[16];
#pragma unroll
    for (int k = 0; k < 16; ++k) hrow[k] = encT[n * 18 + k];   // h row for batch b

    const float x0 = x[b * 72 + t * 3 + 0];
    const float x1 = x[b * 72 + t * 3 + 1];
    const float x2 = x[b * 72 + t * 3 + 2];

    // input attention (per-lane, IN=3); indices into hrow are all compile-time
    float vj[3];
#pragma unroll
    for (int j = 0; j < 3; ++j) {
      float w = a1b[j] + a2b[j] + x0 * a2w[j * 3 + 0] + x1 * a2w[j * 3 + 1] + x2 * a2w[j * 3 + 2];
#pragma unroll
      for (int k = 0; k < 16; ++k) w += hrow[k] * a1w[j * 16 + k];
      vj[j] = ftanh(w);
    }
    float u0 = a3b[0] + vj[0] * a3w[0] + vj[1] * a3w[1] + vj[2] * a3w[2];
    float u1 = a3b[1] + vj[0] * a3w[3] + vj[1] * a3w[4] + vj[2] * a3w[5];
    float u2 = a3b[2] + vj[0] * a3w[6] + vj[1] * a3w[7] + vj[2] * a3w[8];
    const float mxu = fmaxf(u0, fmaxf(u1, u2));
    const float e0 = fexp2(1.44269504f * (u0 - mxu));
    const float e1 = fexp2(1.44269504f * (u1 - mxu));
    const float e2 = fexp2(1.44269504f * (u2 - mxu));
    const float irs = frcp(e0 + e1 + e2);
    const float wx0 = x0 * e0 * irs, wx1 = x1 * e1 * irs, wx2 = x2 * e2 * irs;
    const v2f aWX = hi ? mk2(wx2, 0.f) : mk2(wx0, wx1);

    // GRU gates via WMMA (gate order r,z,n)
    v8f Tr = bc8(b_r), Tz = bc8(b_z), Tic = bc8(b_ic), Thc = bc8(b_hc);
    Tr  = wmma4(aWX, fEIH[0], Tr);
    Tz  = wmma4(aWX, fEIH[1], Tz);
    Tic = wmma4(aWX, fEIH[2], Tic);
#pragma unroll
    for (int c = 0; c < 4; ++c) {
      // A-chunk straight from LDS (ds_load_b64; no dynamic register indexing)
      const v2f aH = *(const v2f*)&encT[n * 18 + 4 * c + 2 * hi];
      Tr  = wmma4(aH, fEHH[0][c], Tr);
      Tz  = wmma4(aH, fEHH[1][c], Tz);
      Thc = wmma4(aH, fEHH[2][c], Thc);
    }

    float* encN = &sENC[(t + 1) * 288];
#pragma unroll
    for (int p = 0; p < 8; ++p) {
      const float r  = sigm(Tr[p]);
      const float z  = sigm(Tz[p]);
      const float nn = ftanh(Tic[p] + r * Thc[p]);
      const float hv = (1.f - z) * nn + z * hC[p];
      hC[p] = hv;
      encN[(p + 8 * hi) * 18 + n] = hv;
    }
    __syncthreads();
  }

  // ============================ DECODER ============================
#pragma unroll 1
  for (int t = 0; t < NT; ++t) {
#pragma unroll
    for (int p = 0; p < 8; ++p) {
      sHD[(p + 8 * hi) * 34 + n]      = hd0[p];
      sHD[(p + 8 * hi) * 34 + 16 + n] = hd1[p];
    }
    __syncthreads();

    // w1 = hd @ l1_w^T + l1_b  (kept as fragments)
    v8f W1a = bc8(bl1a), W1b = bc8(bl1b);
#pragma unroll
    for (int c = 0; c < 8; ++c) {
      const v2f aH = *(const v2f*)&sHD[n * 34 + 4 * c + 2 * hi];
      W1a = wmma4(aH, fL1[0][c], W1a);
      W1b = wmma4(aH, fL1[1][c], W1b);
    }

    // scores a[b,t2] = l3_b + sum_k l3_w[k] * tanh(w1 + w2d[t2]), w2d recomputed via WMMA
#pragma unroll 1
    for (int t2 = 0; t2 < NT; ++t2) {
      const float* e = &sENC[(t2 + 1) * 288];
      v8f W2a = bc8(bl2a), W2b = bc8(bl2b);
#pragma unroll
      for (int c = 0; c < 4; ++c) {
        const v2f aE = *(const v2f*)&e[n * 18 + 4 * c + 2 * hi];
        W2a = wmma4(aE, fL2[0][c], W2a);
        W2b = wmma4(aE, fL2[1][c], W2b);
      }
      float s[8];
#pragma unroll
      for (int p = 0; p < 8; ++p)
        s[p] = l3wa * ftanh(W1a[p] + W2a[p]) + l3wb * ftanh(W1b[p] + W2b[p]);
      // reduce over the 16 columns (lanes within each half)
#pragma unroll
      for (int msk = 1; msk < 16; msk <<= 1) {
#pragma unroll
        for (int p = 0; p < 8; ++p) s[p] += __shfl_xor(s[p], msk, 32);
      }
#pragma unroll
      for (int p = 0; p < 8; ++p)
        if (n == p) sAT[(p + 8 * hi) * 25 + t2] = l3bs + s[p];
    }
    __syncthreads();

    // softmax over t2 (per-lane, batch row n; both halves redundant)
    float av[NT];
    float mx = -1e30f;
#pragma unroll
    for (int tt = 0; tt < NT; ++tt) { av[tt] = sAT[n * 25 + tt]; mx = fmaxf(mx, av[tt]); }
    float ssum = 0.f;
#pragma unroll
    for (int tt = 0; tt < NT; ++tt) {
      av[tt] = fexp2(1.44269504f * (av[tt] - mx));
      ssum += av[tt];
    }
    const float rs = frcp(ssum);

    // context wc[b,j] (lane half covers 8 hidden cols)
    float wc[8];
#pragma unroll
    for (int q = 0; q < 8; ++q) wc[q] = 0.f;
#pragma unroll
    for (int tt = 0; tt < NT; ++tt) {
      const float al = av[tt] * rs;
      const float* e = &sENC[(tt + 1) * 288 + n * 18 + 8 * hi];
#pragma unroll
      for (int q = 0; q < 8; ++q) wc[q] += al * e[q];
    }
#pragma unroll
    for (int q = 0; q < 8; ++q) sY0[n * 22 + 8 * hi + q] = wc[q];
    const float xx0 = x[b * 72 + t * 3 + 0];
    const float xx1 = x[b * 72 + t * 3 + 1];
    const float xx2 = x[b * 72 + t * 3 + 2];
    if (hi) {
      sY0[n * 22 + 16] = xx0; sY0[n * 22 + 17] = xx1;
      sY0[n * 22 + 18] = xx2; sY0[n * 22 + 19] = 0.f;
    }
    __syncthreads();

    // y0 = [wc | x] @ l4_w^T + l4_b   (K = 19 padded to 20)
    v8f Ya = bc8(bl4a), Yb = bc8(bl4b);
#pragma unroll
    for (int c = 0; c < 5; ++c) {
      const v2f aY = *(const v2f*)&sY0[n * 22 + 4 * c + 2 * hi];
      Ya = wmma4(aY, *(const v2f*)&sL4W[n * 22 + 4 * c + 2 * hi], Ya);
      Yb = wmma4(aY, *(const v2f*)&sL4W[(16 + n) * 22 + 4 * c + 2 * hi], Yb);
    }
#pragma unroll
    for (int p = 0; p < 8; ++p) {
      sYT[(p + 8 * hi) * 34 + n]      = Ya[p];
      sYT[(p + 8 * hi) * 34 + 16 + n] = Yb[p];
    }
    __syncthreads();

    // decoder GRU gates (96 wide = 3 gates x 2 tiles), K=32 from y0 and from hd
    v8f Gr0 = bc8(bdr0), Gr1 = bc8(bdr1), Gz0 = bc8(bdz0), Gz1 = bc8(bdz1);
    v8f Gi0 = bc8(bdi0), Gi1 = bc8(bdi1), Gh0 = bc8(bdh0), Gh1 = bc8(bdh1);
#pragma unroll
    for (int c = 0; c < 8; ++c) {
      const int ko = 4 * c + 2 * hi;
      const v2f aY = *(const v2f*)&sYT[n * 34 + ko];
      const v2f aH = *(const v2f*)&sHD[n * 34 + ko];
      Gr0 = wmma4(aY, *(const v2f*)&dwih[(n)      * 32 + ko], Gr0);
      Gr1 = wmma4(aY, *(const v2f*)&dwih[(16 + n) * 32 + ko], Gr1);
      Gz0 = wmma4(aY, *(const v2f*)&dwih[(32 + n) * 32 + ko], Gz0);
      Gz1 = wmma4(aY, *(const v2f*)&dwih[(48 + n) * 32 + ko], Gz1);
      Gi0 = wmma4(aY, *(const v2f*)&dwih[(64 + n) * 32 + ko], Gi0);
      Gi1 = wmma4(aY, *(const v2f*)&dwih[(80 + n) * 32 + ko], Gi1);
      Gr0 = wmma4(aH, *(const v2f*)&dwhh[(n)      * 32 + ko], Gr0);
      Gr1 = wmma4(aH, *(const v2f*)&dwhh[(16 + n) * 32 + ko], Gr1);
      Gz0 = wmma4(aH, *(const v2f*)&dwhh[(32 + n) * 32 + ko], Gz0);
      Gz1 = wmma4(aH, *(const v2f*)&dwhh[(48 + n) * 32 + ko], Gz1);
      Gh0 = wmma4(aH, *(const v2f*)&dwhh[(64 + n) * 32 + ko], Gh0);
      Gh1 = wmma4(aH, *(const v2f*)&dwhh[(80 + n) * 32 + ko], Gh1);
    }
#pragma unroll
    for (int p = 0; p < 8; ++p) {
      const float r0 = sigm(Gr0[p]), z0 = sigm(Gz0[p]);
      const float n0 = ftanh(Gi0[p] + r0 * Gh0[p]);
      hd0[p] = (1.f - z0) * n0 + z0 * hd0[p];
      const float r1 = sigm(Gr1[p]), z1 = sigm(Gz1[p]);
      const float n1 = ftanh(Gi1[p] + r1 * Gh1[p]);
      hd1[p] = (1.f - z1) * n1 + z1 * hd1[p];
    }
    __syncthreads();
  }

  // ---- final fc: out = hd @ fc_w^T + fc_b (OUT=3)
#pragma unroll
  for (int p = 0; p < 8; ++p) {
    sHD[(p + 8 * hi) * 34 + n]      = hd0[p];
    sHD[(p + 8 * hi) * 34 + 16 + n] = hd1[p];
  }
  __syncthreads();
  if (hi == 0) {
    const float* hr = &sHD[n * 34];
#pragma unroll
    for (int j = 0; j < 3; ++j) {
      float o = fcb[j];
#pragma unroll
      for (int k = 0; k < 32; ++k) o += hr[k] * fcw[j * 32 + k];
      out[(base + n) * 3 + j] = o;
    }
  }
}

extern "C" void kernel_launch(void* const* d_in, const int* in_sizes, int n_in,
                              void* d_out, int out_size, void* d_ws, size_t ws_size,
                              hipStream_t stream) {
  (void)in_sizes; (void)n_in; (void)out_size; (void)d_ws; (void)ws_size;
  darnn_kernel<<<dim3(NB / 16), dim3(32), 0, stream>>>(
      (const float*)d_in[0],  (const float*)d_in[1],  (const float*)d_in[2],
      (const float*)d_in[3],  (const float*)d_in[4],  (const float*)d_in[5],
      (const float*)d_in[6],  (const float*)d_in[7],  (const float*)d_in[8],
      (const float*)d_in[9],  (const float*)d_in[10], (const float*)d_in[11],
      (const float*)d_in[12], (const float*)d_in[13], (const float*)d_in[14],
      (const float*)d_in[15], (const float*)d_in[16], (const float*)d_in[17],
      (const float*)d_in[18], (const float*)d_in[19], (const float*)d_in[20],
      (const float*)d_in[21], (const float*)d_in[22], (const float*)d_in[23],
      (const float*)d_in[24], (const float*)d_in[25], (const float*)d_in[26],
      (float*)d_out);
}